// CutsSelector_16037407883356
// MI455X (gfx1250) — compile-verified
//
#include <hip/hip_runtime.h>
#include <hip/hip_bf16.h>
#include <math.h>

typedef __attribute__((ext_vector_type(2))) float v2f;
typedef __attribute__((ext_vector_type(8))) float v8f;

#define NN 20000      // nodes
#define NE 640000     // edges
#define CH 128        // channels

// Workspace layout (float offsets)
#define OFF_P1  0                        // [NN, CH]  x @ g_w[0:128]  (+ g_b folded in)
#define OFF_P2  (NN * CH)                // [NN, CH]  x @ g_w[128:256]
#define OFF_SUM (2 * NN * CH)            // [NN, CH]  scatter sums
#define OFF_CNT (3 * NN * CH)            // [NN]      scatter counts (as float)
#define OFF_V   (3 * NN * CH + NN)       // [256] v1|v2, then [1] folded bias

// ---------------------------------------------------------------- zero sums/cnt
__global__ __launch_bounds__(256) void zero_kernel(float* __restrict__ ws) {
    long long i = (long long)blockIdx.x * blockDim.x + threadIdx.x;
    const long long n = (long long)NN * CH + NN;
    if (i < n) ws[OFF_SUM + i] = 0.0f;
}

// ------------------------------------------- fold f_w/f_b/cls into v1,v2,bias
__global__ __launch_bounds__(256) void fold_kernel(const float* __restrict__ f_w,
                                                   const float* __restrict__ f_b,
                                                   const float* __restrict__ cls_w,
                                                   const float* __restrict__ cls_b,
                                                   float* __restrict__ ws) {
    const int i = threadIdx.x;            // 0..255 -> rows of f_w [256,128]
    float s = 0.0f;
    for (int j = 0; j < CH; ++j) s += f_w[i * CH + j] * cls_w[j];
    ws[OFF_V + i] = s;
    if (i == 0) {
        float b = cls_b[0];
        for (int j = 0; j < CH; ++j) b += f_b[j] * cls_w[j];
        ws[OFF_V + 256] = b;
    }
}

// ------------------------------- node projections P1/P2 via FP32 WMMA 16x16x4
// One wave (32 threads) computes a 16-row x 128-col tile. blockIdx.y selects
// which half of g_w (P1 rows 0..127, P2 rows 128..255).
__global__ __launch_bounds__(32) void proj_kernel(const float* __restrict__ x,
                                                  const float* __restrict__ g_w,
                                                  const float* __restrict__ g_b,
                                                  float* __restrict__ ws) {
    const int L    = threadIdx.x;
    const int ml   = L & 15;          // M for A, N for B/C/D
    const int half = L >> 4;          // selects K pair (A/B) / M+8 (C/D)
    const int row0 = blockIdx.x * 16;
    const int which = blockIdx.y;     // 0 -> P1, 1 -> P2
    const float* W = g_w + which * CH * CH;        // [128,128] row-major (K,N)
    float* P = ws + (which ? OFF_P2 : OFF_P1);

    v8f acc[8];
#pragma unroll
    for (int t = 0; t < 8; ++t) {
        const float binit = which ? 0.0f : g_b[t * 16 + ml];  // fold g_b into P1
#pragma unroll
        for (int v = 0; v < 8; ++v) acc[t][v] = binit;
    }

    const float* xrow = x + (long long)(row0 + ml) * CH;
    for (int kb = 0; kb < CH; kb += 4) {
        const int k = kb + 2 * half;
        v2f a;                                    // A 16x4: lane ml holds K=k,k+1
        a.x = xrow[k];
        a.y = xrow[k + 1];
#pragma unroll
        for (int t = 0; t < 8; ++t) {
            v2f b;                                // B 4x16: lane ml holds col n
            b.x = W[(long long)k * CH + t * 16 + ml];
            b.y = W[(long long)(k + 1) * CH + t * 16 + ml];
            acc[t] = __builtin_amdgcn_wmma_f32_16x16x4_f32(
                false, a, false, b, (short)0, acc[t], false, false);
        }
    }

#pragma unroll
    for (int t = 0; t < 8; ++t)
#pragma unroll
        for (int v = 0; v < 8; ++v)
            P[(long long)(row0 + v + 8 * half) * CH + t * 16 + ml] = acc[t][v];
}

// ------------------------------------------------ edge pass: scatter-add mean
// One wave per edge; each lane handles 4 channels (float4).
__global__ __launch_bounds__(256) void edge_kernel(const long long* __restrict__ ei,
                                                   const float* __restrict__ eattr,
                                                   const float* __restrict__ g_w,
                                                   float* __restrict__ ws) {
    const int lane = threadIdx.x & 31;
    const long long e = (long long)blockIdx.x * 8 + (threadIdx.x >> 5);
    if (e >= NE) return;
    const long long s = ei[e];            // src
    const long long d = ei[NE + e];       // dst
    const float attr = eattr[e];

    const float4 p1 = ((const float4*)(ws + OFF_P1 + d * CH))[lane];
    const float4 p2 = ((const float4*)(ws + OFF_P2 + s * CH))[lane];
    const float4 we = ((const float4*)(g_w + 256 * CH))[lane];   // edge_attr row

    float4 m;
    m.x = p1.x + p2.x + attr * we.x;
    m.y = p1.y + p2.y + attr * we.y;
    m.z = p1.z + p2.z + attr * we.z;
    m.w = p1.w + p2.w + attr * we.w;

    float* sums = ws + OFF_SUM + d * CH + lane * 4;
    atomicAdd(sums + 0, m.x);
    atomicAdd(sums + 1, m.y);
    atomicAdd(sums + 2, m.z);
    atomicAdd(sums + 3, m.w);
    if (lane == 0) atomicAdd(ws + OFF_CNT + d, 1.0f);
}

// ------------------------- node finalize: logit = x.v1 + (sums/max(cnt,1)).v2 + b
__global__ __launch_bounds__(256) void node_kernel(const float* __restrict__ x,
                                                   const float* __restrict__ ws,
                                                   float* __restrict__ out) {
    const int lane = threadIdx.x & 31;
    const int n = blockIdx.x * 8 + (threadIdx.x >> 5);
    if (n >= NN) return;

    const float4 xv = ((const float4*)(x + (long long)n * CH))[lane];
    const float4 sv = ((const float4*)(ws + OFF_SUM + (long long)n * CH))[lane];
    const float4 v1 = ((const float4*)(ws + OFF_V))[lane];
    const float4 v2 = ((const float4*)(ws + OFF_V + CH))[lane];

    float t1 = xv.x * v1.x + xv.y * v1.y + xv.z * v1.z + xv.w * v1.w;
    float t2 = sv.x * v2.x + sv.y * v2.y + sv.z * v2.z + sv.w * v2.w;
#pragma unroll
    for (int off = 16; off > 0; off >>= 1) {
        t1 += __shfl_xor(t1, off, 32);
        t2 += __shfl_xor(t2, off, 32);
    }
    if (lane == 0) {
        const float cnt = ws[OFF_CNT + n];
        const float logit = t1 + t2 / fmaxf(cnt, 1.0f) + ws[OFF_V + 256];
        const float p = 1.0f / (1.0f + expf(-logit));
        out[n]      = (p > 0.5f) ? 1.0f : 0.0f;   // y
        out[NN + n] = p;                          // probs
    }
}

extern "C" void kernel_launch(void* const* d_in, const int* in_sizes, int n_in,
                              void* d_out, int out_size, void* d_ws, size_t ws_size,
                              hipStream_t stream) {
    const float*     x_a   = (const float*)d_in[0];
    const long long* ei    = (const long long*)d_in[1];   // int64 [2, E]
    const float*     eattr = (const float*)d_in[2];
    const float*     g_w   = (const float*)d_in[3];       // [257, 128]
    const float*     g_b   = (const float*)d_in[4];       // [128]
    const float*     f_w   = (const float*)d_in[5];       // [256, 128]
    const float*     f_b   = (const float*)d_in[6];       // [128]
    const float*     cls_w = (const float*)d_in[7];       // [128]
    const float*     cls_b = (const float*)d_in[8];       // [1]
    float* out = (float*)d_out;
    float* ws  = (float*)d_ws;

    // 1) zero sums + counts (ws is poisoned by the harness)
    {
        const long long n = (long long)NN * CH + NN;
        zero_kernel<<<(unsigned)((n + 255) / 256), 256, 0, stream>>>(ws);
    }
    // 2) fold classifier through update MLP: v1, v2, bias
    fold_kernel<<<1, 256, 0, stream>>>(f_w, f_b, cls_w, cls_b, ws);
    // 3) WMMA node projections P1 = x@g_w[0:128] (+g_b), P2 = x@g_w[128:256]
    proj_kernel<<<dim3(NN / 16, 2), 32, 0, stream>>>(x_a, g_w, g_b, ws);
    // 4) edge scatter: sums[dst] += P1[dst]+P2[src]+attr*g_w[256]; cnt[dst]++
    edge_kernel<<<NE / 8, 256, 0, stream>>>(ei, eattr, g_w, ws);
    // 5) finalize: mean, fused update+classifier, sigmoid, threshold
    node_kernel<<<NN / 8, 256, 0, stream>>>(x_a, ws, out);
}